// FEconvNet_periodicU_14121852470125
// MI455X (gfx1250) — compile-verified
//
#include <hip/hip_runtime.h>

typedef __attribute__((ext_vector_type(2))) float v2f;
typedef __attribute__((ext_vector_type(8))) float v8f;

#define NGRID   64
#define NNODE   (NGRID * NGRID * NGRID)          // 262144
#define NTYPES  256
#define PERM_LEN (NNODE + NTYPES * 16)           // padded perm length
#define MAX_TILES (NNODE / 16 + NTYPES)          // 16640 upper bound
#define FILT_LAST (NTYPES * 243 - 1)             // 62207: last valid filter index

// workspace layout (int units)
#define WS_TYPES   0                              // [NNODE]
#define WS_HIST    (NNODE)                        // [256]
#define WS_STARTS  (WS_HIST + NTYPES)             // [257]
#define WS_CURS    (WS_STARTS + NTYPES + 1)       // [256]
#define WS_TOTAL   (WS_CURS + NTYPES)             // [1]
#define WS_PERM    262928                         // [PERM_LEN], 16-aligned

__global__ void k_zero_hist(int* ws) {
    int i = blockIdx.x * blockDim.x + threadIdx.x;
    if (i < NTYPES) ws[WS_HIST + i] = 0;
}

// types[n]: 8-bit occupancy of rho>0.5 over (x..x+1, y..y+1, z..z+1), periodic
__global__ void k_types(const float* __restrict__ rho,
                        const int* __restrict__ tw, int* ws) {
    int n = blockIdx.x * blockDim.x + threadIdx.x;
    if (n >= NNODE) return;
    int x = n >> 12, y = (n >> 6) & 63, z = n & 63;
    int t = 0;
    int i = 0;
    #pragma unroll
    for (int dx = 0; dx < 2; ++dx)
        #pragma unroll
        for (int dy = 0; dy < 2; ++dy)
            #pragma unroll
            for (int dz = 0; dz < 2; ++dz, ++i) {
                int xx = (x + dx) & 63, yy = (y + dy) & 63, zz = (z + dz) & 63;
                float r = rho[(xx << 12) | (yy << 6) | zz];
                if (r > 0.5f) t += (1 << tw[i]);
            }
    t &= (NTYPES - 1);
    ws[WS_TYPES + n] = t;
    atomicAdd(&ws[WS_HIST + t], 1);
}

// serial padded exclusive scan over 256 bins (tiny)
__global__ void k_scan(int* ws) {
    if (blockIdx.x != 0 || threadIdx.x != 0) return;
    int s = 0;
    for (int t = 0; t < NTYPES; ++t) {
        ws[WS_STARTS + t] = s;
        ws[WS_CURS + t]   = s;
        int c = ws[WS_HIST + t];
        s += ((c + 15) >> 4) << 4;   // pad each bucket to a multiple of 16
    }
    ws[WS_STARTS + NTYPES] = s;
    ws[WS_TOTAL] = s;
}

__global__ void k_fill_perm(int* ws) {
    int i = blockIdx.x * blockDim.x + threadIdx.x;
    if (i < PERM_LEN) ws[WS_PERM + i] = -1;
}

__global__ void k_scatter(int* ws) {
    int n = blockIdx.x * blockDim.x + threadIdx.x;
    if (n >= NNODE) return;
    int t = ws[WS_TYPES + n];
    int pos = atomicAdd(&ws[WS_CURS + t], 1);
    ws[WS_PERM + pos] = n;
}

// One wave32 = one tile of 16 same-type nodes.
// D(16x16) = A(16x4: filter rows, M=0..2 meaningful) x B(4x16: Ug columns),
// 21 K-chunks of 4, accumulated in f32.
//
// Key trick: A rows m>=3 and K>=81 may hold garbage — D rows 3..15 are never
// stored, and K>=81 garbage multiplies B entries that are identically zero.
// So all filter loads are unconditional (index clamped in-bounds) and hoisted
// into registers BEFORE the WMMA loop: 21 back-to-back v_wmma with no
// per-chunk loads/waits/exec-mask churn.
__global__ __launch_bounds__(32) void k_wmma(const float* __restrict__ U,
                                             const float* __restrict__ filters,
                                             const int* __restrict__ ws,
                                             float* __restrict__ out) {
    int tile  = blockIdx.x;
    int total = ws[WS_TOTAL];
    if (tile * 16 >= total) return;           // uniform exit

    int lane = threadIdx.x;                   // 0..31
    int m    = lane & 15;
    int half = lane >> 4;                     // 0: K pair {4c,4c+1}, 1: {4c+2,4c+3}

    int n  = ws[WS_PERM + tile * 16 + m];     // this column's node (-1 = padding)
    int n0 = ws[WS_PERM + tile * 16];         // first entry always valid
    int t  = ws[WS_TYPES + n0];               // tile's element type

    // ---- A-fragment preload (unconditional, clamped, fully hoisted) ----
    int mc    = (m < 3) ? m : 2;              // keep addresses near-valid
    int fbase = t * 243 + mc * 81 + 2 * half; // this lane's K offset within row
    float fa[42];
    #pragma unroll
    for (int c = 0; c < 21; ++c) {
        int i0 = fbase + 4 * c;
        int i1 = i0 + 1;
        i0 = (i0 < FILT_LAST) ? i0 : FILT_LAST;   // clamp only engages at K>=81,
        i1 = (i1 < FILT_LAST) ? i1 : FILT_LAST;   // where B columns are zero
        fa[2 * c]     = filters[i0];
        fa[2 * c + 1] = filters[i1];
    }

    // ---- B-fragment gather: this node's Ug[84] (81 real + 3 zero pad) ----
    float ug[84];
    #pragma unroll
    for (int k = 81; k < 84; ++k) ug[k] = 0.0f;
    if (n >= 0) {
        int x = n >> 12, y = (n >> 6) & 63, z = n & 63;
        #pragma unroll
        for (int j = 0; j < 27; ++j) {
            int dx = j / 9 - 1, dy = (j / 3) % 3 - 1, dz = j % 3 - 1;
            int ni = (((x + dx + 64) & 63) << 12) |
                     (((y + dy + 64) & 63) << 6)  |
                     ((z + dz + 64) & 63);
            ug[3 * j + 0] = U[ni * 3 + 0];
            ug[3 * j + 1] = U[ni * 3 + 1];
            ug[3 * j + 2] = U[ni * 3 + 2];
        }
    } else {
        #pragma unroll
        for (int k = 0; k < 81; ++k) ug[k] = 0.0f;
    }

    // ---- 21 back-to-back f32 WMMAs ----
    v8f acc = {};
    #pragma unroll
    for (int c = 0; c < 21; ++c) {
        v2f a, b;
        a.x = fa[2 * c];
        a.y = fa[2 * c + 1];
        // static register indices; uniform-per-lane select (v_cndmask, no spill)
        b.x = half ? ug[4 * c + 2] : ug[4 * c + 0];
        b.y = half ? ug[4 * c + 3] : ug[4 * c + 1];
        acc = __builtin_amdgcn_wmma_f32_16x16x4_f32(
            false, a, false, b, (short)0, acc, false, false);
    }

    // D: VGPR r, lanes 0-15 -> (M=r, N=lane); rows 0..2 are the 3 components
    if (half == 0 && n >= 0) {
        out[n * 3 + 0] = acc[0];
        out[n * 3 + 1] = acc[1];
        out[n * 3 + 2] = acc[2];
    }
}

extern "C" void kernel_launch(void* const* d_in, const int* in_sizes, int n_in,
                              void* d_out, int out_size, void* d_ws, size_t ws_size,
                              hipStream_t stream) {
    (void)in_sizes; (void)n_in; (void)out_size; (void)ws_size;
    const float* U       = (const float*)d_in[0];   // [NNODE,3]
    const float* rho     = (const float*)d_in[1];   // [64,64,64]
    // d_in[2] = nodIdx: deliberately unused (recomputed via shift/AND)
    const float* filters = (const float*)d_in[3];   // [256,3,81]
    const int*   tw      = (const int*)d_in[4];     // [8]
    float* out = (float*)d_out;
    int*   ws  = (int*)d_ws;

    k_zero_hist<<<1, 256, 0, stream>>>(ws);
    k_types<<<NNODE / 256, 256, 0, stream>>>(rho, tw, ws);
    k_scan<<<1, 1, 0, stream>>>(ws);
    k_fill_perm<<<(PERM_LEN + 255) / 256, 256, 0, stream>>>(ws);
    k_scatter<<<NNODE / 256, 256, 0, stream>>>(ws);
    k_wmma<<<MAX_TILES, 32, 0, stream>>>(U, filters, ws, out);
}